// HashGridSDF_50835232915888
// MI455X (gfx1250) — compile-verified
//
#include <hip/hip_runtime.h>
#include <math.h>

#define NLEV 16
#define TABLE_SIZE (1u << 19)
#define HASH_MASK (TABLE_SIZE - 1u)
#define P1 2654435761u
#define P2 805459861u
#define BLK 256

typedef float vf4 __attribute__((ext_vector_type(4)));

struct LevelRes { float r[NLEV]; };

__device__ __forceinline__ float2 lerp2(float2 a, float2 b, float w, float omw) {
    float2 r;
    r.x = a.x * omw + b.x * w;
    r.y = a.y * omw + b.y * w;
    return r;
}

// Gather one float2 from a wave-uniform base + divergent 32-bit byte offset.
// Shapes into: global_load_b64 vD, vOff, s[base] (saddr form, no 64-bit VALU).
__device__ __forceinline__ float2 gather2(const char* __restrict__ base, unsigned byteoff) {
    return *(const float2*)(base + (size_t)byteoff);
}

__global__ __launch_bounds__(BLK) void hashgrid_fwd(
    const float* __restrict__ coords,   // (N,3)
    const float* __restrict__ tables,   // (16, 2^19, 2)
    float* __restrict__ out,            // (N,32)
    LevelRes R, int npts)
{
    __shared__ float sc[3 * BLK];
    const int tid = threadIdx.x;
    const int p0  = blockIdx.x * BLK;

    // --- Coalesced AoS coord staging: 3 async global->LDS b32 copies/thread.
    // Branchless: clamp the global index (duplicate load for tail is harmless).
    const int gbase = p0 * 3;
    const int glim  = npts * 3 - 1;
#pragma unroll
    for (int t = 0; t < 3; ++t) {
        int g = gbase + t * BLK + tid;
        g = (g < glim) ? g : glim;
        unsigned lds_addr = (unsigned)(size_t)&sc[t * BLK + tid];
        unsigned long long gaddr = (unsigned long long)(size_t)(coords + g);
        asm volatile("global_load_async_to_lds_b32 %0, %1, off"
                     :: "v"(lds_addr), "v"(gaddr) : "memory");
    }
    asm volatile("s_wait_asynccnt 0" ::: "memory");
    __syncthreads();

    const int p = p0 + tid;
    if (p >= npts) return;

    const float x = sc[tid * 3 + 0];
    const float y = sc[tid * 3 + 1];
    const float z = sc[tid * 3 + 2];

    float2 o[NLEV];

#pragma unroll
    for (int l = 0; l < NLEV; ++l) {
        const float res = R.r[l];
        // scaled = coords*res ; floor ; frac  (all fp32, matching reference)
        const float sx = x * res, sy = y * res, sz = z * res;
        const float fx = floorf(sx), fy = floorf(sy), fz = floorf(sz);
        const float wx = sx - fx, wy = sy - fy, wz = sz - fz;

        // ci = int((floor+off)*res): products of small ints -> exact in fp32.
        // Hash mod 2^19 in uint32 (valid since 2^19 | 2^32).
        const unsigned hx0 = (unsigned)(fx * res);
        const unsigned hx1 = (unsigned)((fx + 1.0f) * res);
        const unsigned hy0 = (unsigned)(fy * res) * P1;
        const unsigned hy1 = (unsigned)((fy + 1.0f) * res) * P1;
        const unsigned hz0 = (unsigned)(fz * res) * P2;
        const unsigned hz1 = (unsigned)((fz + 1.0f) * res) * P2;

        // Wave-uniform level base (SGPR pair); per-lane 22-bit byte offsets.
        const char* __restrict__ tb =
            (const char*)tables + (size_t)l * (TABLE_SIZE * 8u);

        const float2 f000 = gather2(tb, ((hx0 + hy0 + hz0) & HASH_MASK) << 3);
        const float2 f001 = gather2(tb, ((hx0 + hy0 + hz1) & HASH_MASK) << 3);
        const float2 f010 = gather2(tb, ((hx0 + hy1 + hz0) & HASH_MASK) << 3);
        const float2 f011 = gather2(tb, ((hx0 + hy1 + hz1) & HASH_MASK) << 3);
        const float2 f100 = gather2(tb, ((hx1 + hy0 + hz0) & HASH_MASK) << 3);
        const float2 f101 = gather2(tb, ((hx1 + hy0 + hz1) & HASH_MASK) << 3);
        const float2 f110 = gather2(tb, ((hx1 + hy1 + hz0) & HASH_MASK) << 3);
        const float2 f111 = gather2(tb, ((hx1 + hy1 + hz1) & HASH_MASK) << 3);

        const float omx = 1.0f - wx, omy = 1.0f - wy, omz = 1.0f - wz;
        const float2 c00 = lerp2(f000, f100, wx, omx);
        const float2 c01 = lerp2(f001, f101, wx, omx);
        const float2 c10 = lerp2(f010, f110, wx, omx);
        const float2 c11 = lerp2(f011, f111, wx, omx);
        const float2 c0  = lerp2(c00, c10, wy, omy);
        const float2 c1  = lerp2(c01, c11, wy, omy);
        o[l] = lerp2(c0, c1, wz, omz);
    }

    // 8 x b128 non-temporal stores: keep the 128MB output stream out of L2
    // so the 64MB of hash tables stay resident in the 192MB L2.
    vf4* __restrict__ op = (vf4*)(out + (size_t)p * 32);
#pragma unroll
    for (int q = 0; q < 8; ++q) {
        vf4 v;
        v.x = o[2 * q].x;     v.y = o[2 * q].y;
        v.z = o[2 * q + 1].x; v.w = o[2 * q + 1].y;
        __builtin_nontemporal_store(v, op + q);
    }
}

extern "C" void kernel_launch(void* const* d_in, const int* in_sizes, int n_in,
                              void* d_out, int out_size, void* d_ws, size_t ws_size,
                              hipStream_t stream) {
    const float* coords = (const float*)d_in[0];
    const float* tables = (const float*)d_in[1];
    float* out = (float*)d_out;
    const int npts = in_sizes[0] / 3;

    // Mirror numpy: g = log(512/16)/(16-1) in float64;
    // res_i = float32(floor(16 * exp(i*g)))
    LevelRes R;
    const double g = log(32.0) / 15.0;
    for (int i = 0; i < NLEV; ++i)
        R.r[i] = (float)floor(16.0 * exp((double)i * g));

    const int blocks = (npts + BLK - 1) / BLK;
    hashgrid_fwd<<<blocks, BLK, 0, stream>>>(coords, tables, out, R, npts);
}